// crossCorrelation3D_11630771437795
// MI455X (gfx1250) — compile-verified
//
#include <hip/hip_runtime.h>
#include <cstdint>

// Problem constants (fixed by the reference: B=32, C=1, H=W=512, 9x9 box, pad 4)
#define B_   32
#define H_   512
#define W_   512
#define R_   4
#define TH_  20                 // output rows per tile
#define SR_  (TH_ + 2*R_)       // 28 staged rows (tile + halo)
#define SW_  272                // staged width in floats (68 float4 slots: 8L halo pad + 256 + 8R)
#define SC4_ 68                 // float4 slots per staged row
#define XT_  2                  // x tiles (512 / 256 cols-per-block)
#define YT_  26                 // ceil(512 / TH_)
#define NWG_ (XT_*YT_*B_)       // 1664 partials (divisible by 64 for the WMMA reducer)

typedef float v2f __attribute__((ext_vector_type(2)));
typedef float v8f __attribute__((ext_vector_type(8)));

// 9-tap horizontal sums of I, T, I*I, T*T, I*T read from LDS row pointers.
__device__ __forceinline__ void hsum9(const float* __restrict__ rI,
                                      const float* __restrict__ rT, int c,
                                      float& oI, float& oT, float& oII,
                                      float& oTT, float& oIT) {
  float a[9], b[9];
#pragma unroll
  for (int j = 0; j < 9; ++j) { a[j] = rI[c + j]; b[j] = rT[c + j]; }
  float si = 0.f, st = 0.f, sii = 0.f, stt = 0.f, sit = 0.f;
#pragma unroll
  for (int j = 0; j < 9; ++j) {
    si += a[j];
    st += b[j];
    sii = fmaf(a[j], a[j], sii);
    stt = fmaf(b[j], b[j], stt);
    sit = fmaf(a[j], b[j], sit);
  }
  oI = si; oT = st; oII = sii; oTT = stt; oIT = sit;
}

__global__ __launch_bounds__(256) void cc_tile_kernel(const float* __restrict__ inp,
                                                      const float* __restrict__ tgt,
                                                      float* __restrict__ partial) {
  __shared__ float sI[SR_][SW_];
  __shared__ float sT[SR_][SW_];
  __shared__ float red[256];

  const int tid = threadIdx.x;
  const int bx  = blockIdx.x;           // column strip: 0..1
  const int by  = blockIdx.y;           // row tile:     0..25
  const int bz  = blockIdx.z;           // batch image
  const int x0  = bx * 256;
  const int y0  = by * TH_;
  const size_t imgBase = (size_t)bz * ((size_t)H_ * (size_t)W_);

  // ---- Phase 1: stage tile (with halo) into LDS via async global->LDS DMA ----
  // One float4 slot per loop step; OOB slots are filled with 0 (input) and -1
  // (target: becomes exactly 0 after the t = 0.5*t + 0.5 normalization below,
  //  matching the reference's zero-padding of the *normalized* target).
  for (int s = tid; s < SR_ * SC4_; s += 256) {
    const int r  = s / SC4_;
    const int c4 = s - r * SC4_;
    const int gy = y0 - R_ + r;
    const int gx = x0 - 8 + 4 * c4;
    float* li = &sI[r][4 * c4];
    float* lt = &sT[r][4 * c4];
    if (((unsigned)gy < (unsigned)H_) && (gx >= 0) && (gx + 3 < W_)) {
      const float* gi = inp + imgBase + (size_t)gy * W_ + gx;
      const float* gt = tgt + imgBase + (size_t)gy * W_ + gx;
      unsigned loi = (unsigned)(uintptr_t)li;   // low 32 bits of a generic LDS
      unsigned lot = (unsigned)(uintptr_t)lt;   // pointer == DS byte offset
      asm volatile("global_load_async_to_lds_b128 %0, %1, off"
                   :: "v"(loi), "v"(gi) : "memory");
      asm volatile("global_load_async_to_lds_b128 %0, %1, off"
                   :: "v"(lot), "v"(gt) : "memory");
    } else {
      li[0] = 0.f;  li[1] = 0.f;  li[2] = 0.f;  li[3] = 0.f;
      lt[0] = -1.f; lt[1] = -1.f; lt[2] = -1.f; lt[3] = -1.f;
    }
  }
  asm volatile("s_wait_asynccnt 0" ::: "memory");
  __syncthreads();

  // ---- Phase 2: normalize target in LDS: t = raw*0.5 + 0.5 ----
  {
    float* tf = &sT[0][0];
    for (int s = tid; s < SR_ * SW_; s += 256) tf[s] = fmaf(tf[s], 0.5f, 0.5f);
  }
  __syncthreads();

  // ---- Phase 3: sliding-window 9x9 box sums + per-pixel cc ----
  // Thread owns output column x0+tid (staged col tid+8); leftmost tap = tid+4.
  const int cbase = tid + 4;

  float hI[9], hT[9], hII[9], hTT[9], hIT[9];
  float vI = 0.f, vT = 0.f, vII = 0.f, vTT = 0.f, vIT = 0.f;

#pragma unroll
  for (int k = 0; k < 8; ++k) {        // warm-up: staged rows 0..7
    float nI, nT, nII, nTT, nIT;
    hsum9(&sI[k][0], &sT[k][0], cbase, nI, nT, nII, nTT, nIT);
    hI[k] = nI; hT[k] = nT; hII[k] = nII; hTT[k] = nTT; hIT[k] = nIT;
    vI += nI; vT += nT; vII += nII; vTT += nTT; vIT += nIT;
  }

  float acc = 0.f;
#pragma unroll
  for (int oy = 0; oy < TH_; ++oy) {   // fully unrolled -> %9 indices constant
    float nI, nT, nII, nTT, nIT;
    hsum9(&sI[oy + 8][0], &sT[oy + 8][0], cbase, nI, nT, nII, nTT, nIT);
    vI += nI; vT += nT; vII += nII; vTT += nTT; vIT += nIT;

    const float inv = 1.0f / 81.0f;
    const float Ihat = vI * inv;
    const float That = vT * inv;
    const float cross = vIT - Ihat * vT - That * vI + That * Ihat * 81.0f;
    const float Tvar  = vTT - 2.0f * That * vT + That * That * 81.0f;
    const float Ivar  = vII - 2.0f * Ihat * vI + Ihat * Ihat * 81.0f;
    const float cc    = cross * cross / (Tvar * Ivar + 1e-5f);
    if (y0 + oy < H_) acc += cc;       // uniform branch (last row-tile is partial)

    const int os = oy % 9;             // row leaving the window
    const int ns = (oy + 8) % 9;       // slot freed last iteration
    vI -= hI[os]; vT -= hT[os]; vII -= hII[os]; vTT -= hTT[os]; vIT -= hIT[os];
    hI[ns] = nI; hT[ns] = nT; hII[ns] = nII; hTT[ns] = nTT; hIT[ns] = nIT;
  }

  // ---- Phase 4: block reduction -> one partial per workgroup ----
  red[tid] = acc;
  __syncthreads();
#pragma unroll
  for (int o = 128; o > 0; o >>= 1) {
    if (tid < o) red[tid] += red[tid + o];
    __syncthreads();
  }
  if (tid == 0) {
    const int pidx = (bz * YT_ + by) * XT_ + bx;
    partial[pidx] = red[0];
  }
}

// Single-wave WMMA reduction of the 1664 partials.
// B = all-ones is layout-proof: D = A*1 + C, so sum(D) = 16 * sum(A-values)
// no matter how the 64 loaded values map onto A's (m,k) slots.
__global__ __launch_bounds__(32) void cc_reduce_kernel(const float* __restrict__ partial,
                                                       float* __restrict__ out) {
  const int lane = threadIdx.x;  // 0..31, wave32
  v8f c = {};
  v2f ones; ones[0] = 1.0f; ones[1] = 1.0f;
#pragma unroll
  for (int k = 0; k < NWG_ / 64; ++k) {   // 26 accumulating WMMAs
    v2f a;
    a[0] = partial[k * 64 + 2 * lane];
    a[1] = partial[k * 64 + 2 * lane + 1];
    c = __builtin_amdgcn_wmma_f32_16x16x4_f32(false, a, false, ones,
                                              (short)0, c, false, false);
  }
  float s = c[0] + c[1] + c[2] + c[3] + c[4] + c[5] + c[6] + c[7];
#pragma unroll
  for (int off = 16; off > 0; off >>= 1) s += __shfl_xor(s, off, 32);
  s *= (1.0f / 16.0f);                     // 16 identical columns in D
  if (lane == 0) {
    const float npix = (float)B_ * (float)H_ * (float)W_;
    out[0] = -(s / npix);
  }
}

extern "C" void kernel_launch(void* const* d_in, const int* in_sizes, int n_in,
                              void* d_out, int out_size, void* d_ws, size_t ws_size,
                              hipStream_t stream) {
  const float* inp = (const float*)d_in[0];   // "input"
  const float* tgt = (const float*)d_in[1];   // "target"
  float* out      = (float*)d_out;            // single float
  float* partials = (float*)d_ws;             // NWG_ floats of scratch

  dim3 grid(XT_, YT_, B_);
  cc_tile_kernel<<<grid, 256, 0, stream>>>(inp, tgt, partials);
  cc_reduce_kernel<<<1, 32, 0, stream>>>(partials, out);
}